// GCN_81355270521452
// MI455X (gfx1250) — compile-verified
//
#include <hip/hip_runtime.h>

#define N_NODES  20000
#define N_EDGES  320000
#define N_GRAPHS 64
#define HIDDEN   256
#define STRIP    64      // N columns per workgroup panel
#define KP       258     // padded LDS k-stride (floats), even => 8B-aligned b64 reads

typedef __attribute__((ext_vector_type(2))) float v2f;
typedef __attribute__((ext_vector_type(8))) float v8f;

// ---------------- utility ----------------
__global__ void k_zero(float* __restrict__ p, int n) {
    int i = blockIdx.x * blockDim.x + threadIdx.x;
    if (i < n) p[i] = 0.0f;
}

// ---------------- degrees ----------------
__global__ void k_degrees(const int* __restrict__ src, const int* __restrict__ dst,
                          float* __restrict__ deg_out, float* __restrict__ deg_in) {
    int e = blockIdx.x * blockDim.x + threadIdx.x;
    if (e < N_EDGES) {
        atomicAdd(&deg_out[src[e]], 1.0f);
        atomicAdd(&deg_in[dst[e]], 1.0f);
    }
}

// norms, s = deg_in * norm_out (layer-1 scalar message), graph counts
__global__ void k_norms(const float* __restrict__ deg_in, const float* __restrict__ deg_out,
                        const int* __restrict__ gid,
                        float* __restrict__ norm_in, float* __restrict__ norm_out,
                        float* __restrict__ s, float* __restrict__ gcnt) {
    int n = blockIdx.x * blockDim.x + threadIdx.x;
    if (n < N_NODES) {
        float di = deg_in[n], dq = deg_out[n];
        float ni = rsqrtf(fmaxf(di, 1.0f));
        float no = rsqrtf(fmaxf(dq, 1.0f));
        norm_in[n]  = ni;
        norm_out[n] = no;
        s[n] = di * no;
        atomicAdd(&gcnt[gid[n]], 1.0f);
    }
}

// layer-1 edge aggregation collapsed to scalar: t[dst] += s[src]   (W1 is rank-1)
__global__ void k_scatter1(const int* __restrict__ src, const int* __restrict__ dst,
                           const float* __restrict__ s, float* __restrict__ t) {
    int e = blockIdx.x * blockDim.x + threadIdx.x;
    if (e < N_EDGES) atomicAdd(&t[dst[e]], s[src[e]]);
}

// a[n,c] = relu(t[n]*norm_in[n]*W1[c] + b1[c]) * norm_out[n]   (input to layer-2 GEMM)
__global__ void k_layer1(const float* __restrict__ t, const float* __restrict__ norm_in,
                         const float* __restrict__ norm_out,
                         const float* __restrict__ W1, const float* __restrict__ b1,
                         float* __restrict__ a) {
    int idx = blockIdx.x * blockDim.x + threadIdx.x;
    if (idx < N_NODES * HIDDEN) {
        int n = idx >> 8;
        int c = idx & 255;
        float v = fmaxf(t[n] * norm_in[n] * W1[c] + b1[c], 0.0f);
        a[idx] = v * norm_out[n];
    }
}

// ---------------- layer-2 GEMM: async-to-LDS staging + WMMA ----------------
// x2 = a @ W2, (20000x256)@(256x256), fp32 via V_WMMA_F32_16X16X4_F32.
// Block = 8 waves sharing one 64-column B panel of W2 staged into LDS with
// GLOBAL_LOAD_ASYNC_TO_LDS_B32 (per-lane LDS dest => transposing copy).
// LDS layout: Bs[n*KP + k]  (k-contiguous per column) so each B fragment pair
// (k, k+1) is one 8-byte LDS read.
__global__ void __launch_bounds__(256) k_gemm_wmma(const float* __restrict__ A,
                                                   const float* __restrict__ B,
                                                   float* __restrict__ C) {
    __shared__ float Bs[STRIP * KP];   // 64 cols x 256 k (padded) = 66048 B

    const int tid = threadIdx.x;
    const int n0  = (blockIdx.x & 3) * STRIP;   // 4 N-strips
    const int mg  = blockIdx.x >> 2;            // M-group of 8 tiles

    // ---- stage W2[:, n0:n0+64] into LDS (transposed) via async loads ----
    {
        const int n   = tid & 63;               // panel column
        const int kb0 = (tid >> 6) << 6;        // k block: 0,64,128,192
        const float* g = B + (size_t)kb0 * HIDDEN + n0 + n;
        unsigned lds = (unsigned)(uintptr_t)&Bs[n * KP + kb0];
#pragma unroll 8
        for (int i = 0; i < 64; ++i) {
            asm volatile("global_load_async_to_lds_b32 %0, %1, off"
                         :: "v"(lds), "v"(g) : "memory");
            lds += 4;
            g   += HIDDEN;                      // next k row of W2
        }
        asm volatile("s_wait_asynccnt 0" ::: "memory");
    }
    __syncthreads();

    const int wave = tid >> 5;
    const int lane = tid & 31;
    const int m0   = (mg * 8 + wave) * 16;
    if (m0 >= N_NODES) return;                  // after the barrier: safe

    const int half = lane >> 4;                 // 0 or 1
    const int sub  = lane & 15;
    const int kb   = half * 2;                  // K offset within 4-wide step

    const float* arow = A + (size_t)(m0 + sub) * HIDDEN + kb;
    const float* bs0  = &Bs[(sub +  0) * KP + kb];
    const float* bs1  = &Bs[(sub + 16) * KP + kb];
    const float* bs2  = &Bs[(sub + 32) * KP + kb];
    const float* bs3  = &Bs[(sub + 48) * KP + kb];

    v8f c0 = {}, c1 = {}, c2 = {}, c3 = {};
    for (int k = 0; k < HIDDEN; k += 4) {
        v2f af = *(const v2f*)(arow + k);       // A[m0+sub][k+kb .. k+kb+1]
        v2f b0 = *(const v2f*)(bs0 + k);        // ds_load_b64 fragments
        v2f b1 = *(const v2f*)(bs1 + k);
        v2f b2 = *(const v2f*)(bs2 + k);
        v2f b3 = *(const v2f*)(bs3 + k);
        c0 = __builtin_amdgcn_wmma_f32_16x16x4_f32(false, af, false, b0, (short)0, c0, false, false);
        c1 = __builtin_amdgcn_wmma_f32_16x16x4_f32(false, af, false, b1, (short)0, c1, false, false);
        c2 = __builtin_amdgcn_wmma_f32_16x16x4_f32(false, af, false, b2, (short)0, c2, false, false);
        c3 = __builtin_amdgcn_wmma_f32_16x16x4_f32(false, af, false, b3, (short)0, c3, false, false);
    }

#pragma unroll
    for (int r = 0; r < 8; ++r) {
        int row = m0 + r + 8 * half;
        float* cp = C + (size_t)row * HIDDEN + n0 + sub;
        cp[0]  = c0[r];
        cp[16] = c1[r];
        cp[32] = c2[r];
        cp[48] = c3[r];
    }
}

// layer-2 propagation: agg[dst, :] += x2[src, :]  (L2-resident, float4 gather + fp32 atomics)
__global__ void k_scatter256(const int* __restrict__ src, const int* __restrict__ dst,
                             const float* __restrict__ x, float* __restrict__ agg) {
    int tid = blockIdx.x * blockDim.x + threadIdx.x;   // N_EDGES * 64 threads
    int e = tid >> 6;
    int q = (tid & 63) << 2;                            // channel base (quad)
    if (e < N_EDGES) {
        const float4 v = *(const float4*)(x + (size_t)src[e] * HIDDEN + q);
        float* ap = agg + (size_t)dst[e] * HIDDEN + q;
        atomicAdd(ap + 0, v.x);
        atomicAdd(ap + 1, v.y);
        atomicAdd(ap + 2, v.z);
        atomicAdd(ap + 3, v.w);
    }
}

// h2 = relu(agg*norm_in + b2); pool into gsum[graph]
__global__ void k_pool(const float* __restrict__ agg, const float* __restrict__ norm_in,
                       const int* __restrict__ gid, const float* __restrict__ b2,
                       float* __restrict__ gsum) {
    int idx = blockIdx.x * blockDim.x + threadIdx.x;
    if (idx < N_NODES * HIDDEN) {
        int n = idx >> 8;
        int c = idx & 255;
        float v = fmaxf(agg[idx] * norm_in[n] + b2[c], 0.0f);
        atomicAdd(&gsum[gid[n] * HIDDEN + c], v);
    }
}

// out[g, j] = (gsum[g,:] @ Wc[:,j]) / max(gcnt[g],1) + bc[j]
__global__ void k_head(const float* __restrict__ gsum, const float* __restrict__ gcnt,
                       const float* __restrict__ Wc, const float* __restrict__ bc,
                       float* __restrict__ out) {
    int t = threadIdx.x;                // 128 threads
    int g = t >> 1, j = t & 1;
    const float* row = gsum + g * HIDDEN;
    float acc = 0.0f;
    for (int c = 0; c < HIDDEN; ++c) acc += row[c] * Wc[c * 2 + j];
    out[g * 2 + j] = acc / fmaxf(gcnt[g], 1.0f) + bc[j];
}

extern "C" void kernel_launch(void* const* d_in, const int* in_sizes, int n_in,
                              void* d_out, int out_size, void* d_ws, size_t ws_size,
                              hipStream_t stream) {
    (void)in_sizes; (void)n_in; (void)out_size; (void)ws_size;
    const int*   src = (const int*)d_in[0];
    const int*   dst = (const int*)d_in[1];
    const int*   gid = (const int*)d_in[2];
    const float* W1  = (const float*)d_in[3];
    const float* b1  = (const float*)d_in[4];
    const float* W2  = (const float*)d_in[5];
    const float* b2  = (const float*)d_in[6];
    const float* Wc  = (const float*)d_in[7];
    const float* bc  = (const float*)d_in[8];
    float* out = (float*)d_out;

    // workspace layout (floats); total ~10.38M floats = 41.5 MB
    float* ws       = (float*)d_ws;
    float* deg_in   = ws;
    float* deg_out  = deg_in  + N_NODES;
    float* norm_in  = deg_out + N_NODES;
    float* norm_out = norm_in + N_NODES;
    float* sv       = norm_out + N_NODES;
    float* tv       = sv + N_NODES;
    float* gsum     = tv + N_NODES;
    float* gcnt     = gsum + N_GRAPHS * HIDDEN;
    float* bufA     = gcnt + N_GRAPHS;                 // layer-1 act, reused as layer-2 agg
    float* bufX     = bufA + (size_t)N_NODES * HIDDEN; // GEMM output x2

    const int T = 256;
    const int smallN = 6 * N_NODES + N_GRAPHS * HIDDEN + N_GRAPHS;
    const int bigN   = N_NODES * HIDDEN;

    k_zero<<<(smallN + T - 1) / T, T, 0, stream>>>(ws, smallN);
    k_degrees<<<(N_EDGES + T - 1) / T, T, 0, stream>>>(src, dst, deg_out, deg_in);
    k_norms<<<(N_NODES + T - 1) / T, T, 0, stream>>>(deg_in, deg_out, gid,
                                                     norm_in, norm_out, sv, gcnt);
    k_scatter1<<<(N_EDGES + T - 1) / T, T, 0, stream>>>(src, dst, sv, tv);
    k_layer1<<<(bigN + T - 1) / T, T, 0, stream>>>(tv, norm_in, norm_out, W1, b1, bufA);

    // M-groups of 8 tiles: ceil(1250/8)=157; x4 N-strips => 628 blocks of 8 waves
    k_gemm_wmma<<<157 * 4, T, 0, stream>>>(bufA, W2, bufX);

    k_zero<<<(bigN + T - 1) / T, T, 0, stream>>>(bufA, bigN);      // bufA now agg2
    k_scatter256<<<(N_EDGES * 64 + T - 1) / T, T, 0, stream>>>(src, dst, bufX, bufA);
    k_pool<<<(bigN + T - 1) / T, T, 0, stream>>>(bufA, norm_in, gid, b2, gsum);
    k_head<<<1, 2 * N_GRAPHS, 0, stream>>>(gsum, gcnt, Wc, bc, out);
}